// SeparableWeightedConv_37572373905796
// MI455X (gfx1250) — compile-verified
//
#include <hip/hip_runtime.h>
#include <hip/hip_bf16.h>
#include <cstdint>

// Problem constants (match reference)
#define BB   8
#define CC   128
#define NN   8192      // N == L (pad=4, K=9, stride=1)
#define KK   9
#define PADW 4

#define TILE_L   1024          // l-values per block
#define THREADS  256           // 8 wave32
#define OUT_PT   4             // outputs per thread
#define LDSN     1032          // TILE_L + 8 halo
#define CH_PER_BLOCK 16        // channels per block -> 8 chunks

typedef uint32_t u32x4 __attribute__((ext_vector_type(4)));
typedef uint32_t u32x8 __attribute__((ext_vector_type(8)));
typedef float    f32x4 __attribute__((ext_vector_type(4)));

__device__ __forceinline__ uint32_t lds_off_u32(const void* p) {
    // generic pointer to LDS: low 32 bits = wave-relative LDS byte offset
    return (uint32_t)(uintptr_t)p;
}

__device__ __forceinline__ void async_ld_b128(uint32_t lds_off, const float* gptr) {
    uint64_t ga = (uint64_t)(uintptr_t)gptr;
    asm volatile("global_load_async_to_lds_b128 %0, %1, off"
                 :: "v"(lds_off), "v"(ga) : "memory");
}

__device__ __forceinline__ void wait_async0() {
    asm volatile("s_wait_asynccnt 0" ::: "memory");
}

// One 2-D TDM descriptor load: rows x tile0 elements (f32), row stride in elements.
// tensor_dim0 = dim0_len => elements past it read as ZERO (free right-edge padding).
__device__ __forceinline__ void tdm_load_2d(uint32_t lds_byte_off, uint64_t gaddr,
                                            uint32_t dim0_len, uint32_t rows,
                                            uint32_t tile0, uint32_t tile1,
                                            uint32_t row_stride_elems) {
    u32x4 g0;
    g0.x = 1u;                                             // count=1 (valid), user mode
    g0.y = lds_byte_off;                                   // lds_addr (bytes)
    g0.z = (uint32_t)gaddr;                                // global_addr[31:0]
    g0.w = ((uint32_t)(gaddr >> 32) & 0x01FFFFFFu)         // global_addr[56:32]
           | 0x80000000u;                                  // type=2 ("image") in [127:126]
    u32x8 g1;
    g1.s0 = 2u << 16;                                      // data_size=2 (4 bytes)
    g1.s1 = (dim0_len & 0xFFFFu) << 16;                    // tensor_dim0[15:0] @ bit48
    g1.s2 = (dim0_len >> 16) | ((rows & 0xFFFFu) << 16);   // tensor_dim0[31:16] | tensor_dim1[15:0]
    g1.s3 = (rows >> 16) | (tile0 << 16);                  // tensor_dim1[31:16] | tile_dim0
    g1.s4 = tile1;                                         // tile_dim1 | tile_dim2(=0)<<16
    g1.s5 = row_stride_elems;                              // tensor_dim0_stride[31:0]
    g1.s6 = 0u;                                            // stride0[47:32] | stride1[15:0]
    g1.s7 = 0u;                                            // stride1[47:16]
    asm volatile("tensor_load_to_lds %0, %1" :: "s"(g0), "s"(g1) : "memory");
}

__global__ __launch_bounds__(THREADS)
void swconv_kernel(const float* __restrict__ x,
                   const float* __restrict__ coords,
                   const float* __restrict__ sigma,
                   const float* __restrict__ weight,
                   float* __restrict__ out) {
    __shared__ __align__(16) float sx[CH_PER_BLOCK][LDSN];  // 66 KB: 16 x-rows
    __shared__ __align__(16) float sc[3][LDSN];             // 12 KB: coords rows

    const int tid    = threadIdx.x;
    const int tile   = blockIdx.x;       // 0..7
    const int b      = blockIdx.y;       // 0..7
    const int cchunk = blockIdx.z;       // 0..7
    const int l0     = tile * TILE_L;
    const int c0     = cchunk * CH_PER_BLOCK;
    const int start  = (l0 == 0) ? 0 : (l0 - PADW);   // global index of LDS slot 0

    // ---- 1) wave 0 kicks one TDM for all 16 x-rows (overlaps with coords staging) ---
    if (tid < 32) {
        uint64_t ga = (uint64_t)(uintptr_t)(x + ((size_t)(b * CC + c0)) * NN + start);
        tdm_load_2d(lds_off_u32(&sx[0][0]), ga,
                    (uint32_t)(NN - start),      // OOB (right edge) reads return zero
                    CH_PER_BLOCK, LDSN, CH_PER_BLOCK, NN);
    }

    // ---- 2) all lanes async-stage the 3 coords rows (b128 per lane) ------------
    #pragma unroll
    for (int r = 0; r < 3; ++r) {
        const float* crow = coords + ((size_t)b * 3 + r) * NN;
        int p0 = start + 4 * tid;                     // slots 0..1023
        p0 = p0 > NN - 4 ? NN - 4 : p0;               // clamp (masked via g below)
        async_ld_b128(lds_off_u32(&sc[r][4 * tid]), crow + p0);
        const int i2 = 1024 + 4 * (tid & 1);          // tail slots 1024..1031 (dup ok)
        int p2 = start + i2;
        p2 = p2 > NN - 4 ? NN - 4 : p2;
        async_ld_b128(lds_off_u32(&sc[r][i2]), crow + p2);
    }

    wait_async0();                                      // this wave's async DMA done
    if (tid < 32) __builtin_amdgcn_s_wait_tensorcnt(0); // wave0: TDM done
    __syncthreads();                                    // all DMA visible to block

    // ---- 3) distance weights (zero-pad folded into g) --------------------------
    const float inv_sigma = 1.0f / sigma[0];
    const int base = 4 * tid + (l0 - start) - PADW;   // slot of (l=l0+4*tid, k=0)

    float cw[3][12];
    if (base >= 0) {
        #pragma unroll
        for (int r = 0; r < 3; ++r) {
            const f32x4* sp = (const f32x4*)&sc[r][base];
            f32x4 a = sp[0], bq = sp[1], cq = sp[2];
            #pragma unroll
            for (int m = 0; m < 4; ++m) {
                cw[r][m]     = a[m];
                cw[r][m + 4] = bq[m];
                cw[r][m + 8] = cq[m];
            }
        }
    } else {                                          // only tid 0 of first tile
        #pragma unroll
        for (int r = 0; r < 3; ++r)
            #pragma unroll
            for (int m = 0; m < 12; ++m) {
                int s = base + m; s = s < 0 ? 0 : s;
                cw[r][m] = sc[r][s];
            }
    }

    float g[OUT_PT][KK];
    #pragma unroll
    for (int j = 0; j < OUT_PT; ++j) {
        const float cx = cw[0][j + PADW];             // center tap (always in range)
        const float cy = cw[1][j + PADW];
        const float cz = cw[2][j + PADW];
        #pragma unroll
        for (int k = 0; k < KK; ++k) {
            const int p = l0 + 4 * tid + j + k - PADW;
            const bool in = (p >= 0) && (p < NN);
            const float dx = cw[0][j + k] - cx;
            const float dy = cw[1][j + k] - cy;
            const float dz = cw[2][j + k] - cz;
            const float sq = dx * dx + dy * dy + dz * dz;
            const float dw = fmaxf(1.0f - sqrtf(sq) * inv_sigma, 0.0f);
            g[j][k] = in ? dw : 0.0f;                 // zero-padding folded in
        }
    }

    // ---- 4) channel loop: pure LDS reads + FMA + coalesced NT float4 store -----
    #pragma unroll
    for (int ci = 0; ci < CH_PER_BLOCK; ++ci) {
        const int c = c0 + ci;

        float wk[KK];
        #pragma unroll
        for (int k = 0; k < KK; ++k) wk[k] = weight[c * KK + k];  // uniform -> s_load

        float v[12];
        if (base >= 0) {
            const f32x4* sp = (const f32x4*)&sx[ci][base];        // 3x ds_load_b128
            f32x4 a = sp[0], bq = sp[1], cq = sp[2];
            #pragma unroll
            for (int m = 0; m < 4; ++m) {
                v[m]     = a[m];
                v[m + 4] = bq[m];
                v[m + 8] = cq[m];
            }
        } else {
            #pragma unroll
            for (int m = 0; m < 12; ++m) {
                int s = base + m; s = s < 0 ? 0 : s;
                v[m] = sx[ci][s];
            }
        }

        f32x4 o;
        #pragma unroll
        for (int j = 0; j < OUT_PT; ++j) {
            float acc = 0.0f;
            #pragma unroll
            for (int k = 0; k < KK; ++k)
                acc = fmaf(v[j + k] * g[j][k], wk[k], acc);
            o[j] = acc;
        }
        f32x4* optr = (f32x4*)(out + ((size_t)(b * CC + c)) * NN + l0 + 4 * tid);
        __builtin_nontemporal_store(o, optr);         // streaming write-once output
    }
}

extern "C" void kernel_launch(void* const* d_in, const int* in_sizes, int n_in,
                              void* d_out, int out_size, void* d_ws, size_t ws_size,
                              hipStream_t stream) {
    const float* x      = (const float*)d_in[0];   // (B, C, N, 1)
    const float* coords = (const float*)d_in[1];   // (B, 3, N, 1)
    const float* sigma  = (const float*)d_in[2];   // (1,)
    const float* weight = (const float*)d_in[3];   // (C, 1, K)
    float* out = (float*)d_out;                    // (B, C, L)

    dim3 grid(NN / TILE_L, BB, CC / CH_PER_BLOCK); // (8, 8, 8) = 512 blocks
    dim3 block(THREADS);
    swconv_kernel<<<grid, block, 0, stream>>>(x, coords, sigma, weight, out);
}